// GNNBlock_16655883174661
// MI455X (gfx1250) — compile-verified
//
#include <hip/hip_runtime.h>
#include <hip/hip_bf16.h>
#include <math.h>

typedef __attribute__((ext_vector_type(2))) float v2f;
typedef __attribute__((ext_vector_type(8))) float v8f;

#define IN_CH   768
#define OUT_CH  64
#define NEG_SLOPE 0.2f

// -------------------------------------------------------------------------
// Fused triple GEMM: one wave computes a 16-row tile of all three outputs.
//   x_l  = node @ W_l  + b_l
//   x_r  = node @ W_r  + b_r
//   out  = node @ W_lin + b_lin + gnn_bias   (scatter target, relu'd later)
// Uses V_WMMA_F32_16X16X4_F32 (fp32 WMMA), 12 accumulators per wave
// (3 matrices x 4 column tiles of the 64-wide output), 192 k-steps.
// -------------------------------------------------------------------------
__global__ __launch_bounds__(256) void gemm3_wmma_kernel(
    const float* __restrict__ node,
    const float* __restrict__ Wl,  const float* __restrict__ bl,
    const float* __restrict__ Wr,  const float* __restrict__ br,
    const float* __restrict__ Wlin, const float* __restrict__ blin,
    const float* __restrict__ gbias,
    float* __restrict__ xl, float* __restrict__ xr, float* __restrict__ outlin,
    int nTiles)
{
    const int wave = blockIdx.x * 8 + (threadIdx.x >> 5);
    if (wave >= nTiles) return;
    const int lane  = threadIdx.x & 31;
    const int lm    = lane & 15;            // row within tile (A), col within tile (B)
    const int khalf = (lane >> 4) << 1;     // 0 for lanes 0-15, 2 for lanes 16-31
    const size_t row0 = (size_t)wave * 16;

    const float* __restrict__ W[3] = { Wl, Wr, Wlin };
    v8f acc[3][4] = {};

    const float* aRow = node + (row0 + lm) * IN_CH + khalf;

    for (int k0 = 0; k0 < IN_CH; k0 += 4) {
        // A fragment: lane holds A[m][k0+khalf], A[m][k0+khalf+1]
        v2f a = *(const v2f*)(aRow + k0);
#pragma unroll
        for (int w = 0; w < 3; ++w) {
            const float* Wk0 = W[w] + (size_t)(k0 + khalf) * OUT_CH;
#pragma unroll
            for (int ct = 0; ct < 4; ++ct) {
                const int col = ct * 16 + lm;
                v2f b;
                b.x = Wk0[col];             // B[k0+khalf    ][col]
                b.y = Wk0[OUT_CH + col];    // B[k0+khalf + 1][col]
                acc[w][ct] = __builtin_amdgcn_wmma_f32_16x16x4_f32(
                    false, a, false, b, (short)0, acc[w][ct], false, false);
            }
        }
    }

    // C/D layout: lanes 0-15 -> M = r, N = lane; lanes 16-31 -> M = r+8, N = lane-16
    const int mOff = (lane < 16) ? 0 : 8;
#pragma unroll
    for (int ct = 0; ct < 4; ++ct) {
        const int col = ct * 16 + lm;
        const float bias_l   = bl[col];
        const float bias_r   = br[col];
        const float bias_lin = blin[col] + gbias[col];
#pragma unroll
        for (int r = 0; r < 8; ++r) {
            const size_t idx = (row0 + r + mOff) * OUT_CH + col;
            xl[idx]     = acc[0][ct][r] + bias_l;
            xr[idx]     = acc[1][ct][r] + bias_r;
            outlin[idx] = acc[2][ct][r] + bias_lin;
        }
    }
}

// -------------------------------------------------------------------------
// Init segment-max to -inf and segment-sum to 0.
// -------------------------------------------------------------------------
__global__ __launch_bounds__(256) void init_kernel(float* __restrict__ mmax,
                                                   float* __restrict__ denom, int N)
{
    int i = blockIdx.x * 256 + threadIdx.x;
    if (i < N) {
        mmax[i]  = -__builtin_inff();
        denom[i] = 0.0f;
    }
}

// float atomic max via sign-split int/uint punning (always compiles; lowers to
// global_atomic_max_i32 / global_atomic_min_u32, correct for IEEE ordering).
__device__ inline void atomicMaxF(float* addr, float val) {
    if (val >= 0.0f) atomicMax((int*)addr, __float_as_int(val));
    else             atomicMin((unsigned int*)addr, __float_as_uint(val));
}

// -------------------------------------------------------------------------
// Edge scores: e = att . leaky_relu(x_l[src] + x_r[dst]); segment max over dst.
// One wave per edge, 2 channels per lane (float2).  Edges >= E are self loops.
// -------------------------------------------------------------------------
__global__ __launch_bounds__(256) void edge_score_kernel(
    const long long* __restrict__ ei, int E, int N,
    const float* __restrict__ xl, const float* __restrict__ xr,
    const float* __restrict__ att,
    float* __restrict__ escore, float* __restrict__ mmax)
{
    const int e    = blockIdx.x * 8 + (threadIdx.x >> 5);
    const int lane = threadIdx.x & 31;
    const int ET   = E + N;
    if (e >= ET) return;

    int src, dst;
    if (e < E) { src = (int)ei[e]; dst = (int)ei[(size_t)E + e]; }
    else       { src = dst = e - E; }

    v2f a = *(const v2f*)(xl + (size_t)src * OUT_CH + lane * 2);
    v2f b = *(const v2f*)(xr + (size_t)dst * OUT_CH + lane * 2);
    float tx = a.x + b.x;
    float ty = a.y + b.y;
    tx = (tx > 0.0f) ? tx : tx * NEG_SLOPE;
    ty = (ty > 0.0f) ? ty : ty * NEG_SLOPE;
    v2f av = *(const v2f*)(att + lane * 2);
    float partial = av.x * tx + av.y * ty;

#pragma unroll
    for (int off = 16; off > 0; off >>= 1)
        partial += __shfl_xor(partial, off, 32);

    if (lane == 0) {
        escore[e] = partial;
        atomicMaxF(&mmax[dst], partial);
    }
}

// -------------------------------------------------------------------------
// p = exp(e - m[dst]); denom[dst] += p (segment sum).  Thread per edge;
// p overwrites escore in place.
// -------------------------------------------------------------------------
__global__ __launch_bounds__(256) void edge_exp_kernel(
    const long long* __restrict__ ei, int E, int N,
    float* __restrict__ escore, const float* __restrict__ mmax,
    float* __restrict__ denom)
{
    const int e  = blockIdx.x * 256 + threadIdx.x;
    const int ET = E + N;
    if (e >= ET) return;
    const int dst = (e < E) ? (int)ei[(size_t)E + e] : (e - E);
    const float p = __expf(escore[e] - mmax[dst]);
    escore[e] = p;
    atomicAdd(&denom[dst], p);
}

// -------------------------------------------------------------------------
// Weighted scatter: out[dst] += (p/denom[dst]) * x_l[src].  Wave per edge,
// 2 channels per lane, distinct addresses per lane -> conflict-free atomics.
// -------------------------------------------------------------------------
__global__ __launch_bounds__(256) void edge_scatter_kernel(
    const long long* __restrict__ ei, int E, int N,
    const float* __restrict__ escore, const float* __restrict__ denom,
    const float* __restrict__ xl, float* __restrict__ out)
{
    const int e    = blockIdx.x * 8 + (threadIdx.x >> 5);
    const int lane = threadIdx.x & 31;
    const int ET   = E + N;
    if (e >= ET) return;

    int src, dst;
    if (e < E) { src = (int)ei[e]; dst = (int)ei[(size_t)E + e]; }
    else       { src = dst = e - E; }

    const float alpha = escore[e] / denom[dst];
    v2f xv = *(const v2f*)(xl + (size_t)src * OUT_CH + lane * 2);
    float* p = out + (size_t)dst * OUT_CH + lane * 2;
    atomicAdd(p,     alpha * xv.x);
    atomicAdd(p + 1, alpha * xv.y);
}

// -------------------------------------------------------------------------
// Final ReLU in place.
// -------------------------------------------------------------------------
__global__ __launch_bounds__(256) void relu_kernel(float* __restrict__ out, int n)
{
    int i = blockIdx.x * 256 + threadIdx.x;
    if (i < n) out[i] = fmaxf(out[i], 0.0f);
}

// -------------------------------------------------------------------------
extern "C" void kernel_launch(void* const* d_in, const int* in_sizes, int n_in,
                              void* d_out, int out_size, void* d_ws, size_t ws_size,
                              hipStream_t stream)
{
    const float*     node = (const float*)d_in[0];
    const long long* ei   = (const long long*)d_in[1];   // int64 edge_index [2, E]
    const float*     Wl   = (const float*)d_in[2];
    const float*     bl   = (const float*)d_in[3];
    const float*     Wr   = (const float*)d_in[4];
    const float*     br   = (const float*)d_in[5];
    const float*     att  = (const float*)d_in[6];
    const float*     gb   = (const float*)d_in[7];
    const float*     Wlin = (const float*)d_in[8];
    const float*     blin = (const float*)d_in[9];
    float* out = (float*)d_out;

    const int N  = in_sizes[0] / IN_CH;   // 100000
    const int E  = in_sizes[1] / 2;       // 3200000
    const int ET = E + N;                 // + self loops

    // workspace layout (all fully rewritten every call)
    float* xl    = (float*)d_ws;                 // N*64
    float* xr    = xl + (size_t)N * OUT_CH;      // N*64
    float* esc   = xr + (size_t)N * OUT_CH;      // ET
    float* mmax  = esc + ET;                     // N
    float* denom = mmax + N;                     // N

    // 1) fused triple GEMM (WMMA fp32); out gets linear branch + biases
    const int nTiles = (N + 15) / 16;            // 6250, exact
    gemm3_wmma_kernel<<<(nTiles + 7) / 8, 256, 0, stream>>>(
        node, Wl, bl, Wr, br, Wlin, blin, gb, xl, xr, out, nTiles);

    // 2) init segment max / sum
    init_kernel<<<(N + 255) / 256, 256, 0, stream>>>(mmax, denom, N);

    // 3) edge scores + segment max   (wave per edge)
    edge_score_kernel<<<(ET + 7) / 8, 256, 0, stream>>>(ei, E, N, xl, xr, att, esc, mmax);

    // 4) exp + segment sum           (thread per edge)
    edge_exp_kernel<<<(ET + 255) / 256, 256, 0, stream>>>(ei, E, N, esc, mmax, denom);

    // 5) weighted scatter-add        (wave per edge)
    edge_scatter_kernel<<<(ET + 7) / 8, 256, 0, stream>>>(ei, E, N, esc, denom, xl, out);

    // 6) final ReLU
    const int total = N * OUT_CH;
    relu_kernel<<<(total + 255) / 256, 256, 0, stream>>>(out, total);
}